// SAGEConv_4363686772846
// MI455X (gfx1250) — compile-verified
//
#include <hip/hip_runtime.h>

typedef __attribute__((ext_vector_type(2))) float v2f;
typedef __attribute__((ext_vector_type(8))) float v8f;

#define LDSW 66  // padded W^T row stride: keeps 8B alignment, spreads LDS banks

// ---------------- zero workspace ----------------
__global__ __launch_bounds__(256) void zero_ws_kernel(float* __restrict__ p, long n) {
  long i = (long)blockIdx.x * blockDim.x + threadIdx.x;
  if (i < n) p[i] = 0.0f;
}

__device__ __forceinline__ void atomAddF(float* p, float v) {
  unsafeAtomicAdd(p, v);  // -> global_atomic_add_f32 (no return, STOREcnt)
}

// ---------------- edge scatter: sum[dst] += x[src], deg[dst] += 1 ----------------
// 16 lanes per edge, each lane moves one float4 of the 64-float feature row.
__global__ __launch_bounds__(256) void sage_scatter_kernel(
    const float* __restrict__ x, const long long* __restrict__ ei,
    float* __restrict__ sum, float* __restrict__ deg, long E) {
  long tid = (long)blockIdx.x * blockDim.x + threadIdx.x;
  long e = tid >> 4;
  if (e >= E) return;
  int q = (int)(tid & 15);
  long src = (long)ei[e];
  long dst = (long)ei[E + e];
  const float4 v = *((const float4*)(x + src * 64) + q);  // global_load_b128
  float* s = sum + dst * 64 + (long)q * 4;
  atomAddF(s + 0, v.x);
  atomAddF(s + 1, v.y);
  atomAddF(s + 2, v.z);
  atomAddF(s + 3, v.w);
  if (q == 0) atomAddF(deg + dst, 1.0f);
}

// ---------------- fused mean-normalize + dual GEMM via FP32 WMMA ----------------
// out[r,:] = (sum[r,:]/max(deg[r],1)) @ W_l + x[r,:] @ W_r
// One wave handles 16 rows x 64 cols. Block = 8 waves = 128 rows.
__global__ __launch_bounds__(256) void sage_gemm_kernel(
    const float* __restrict__ x, const float* __restrict__ Wl,
    const float* __restrict__ Wr, const float* __restrict__ sum,
    const float* __restrict__ deg, float* __restrict__ out, int nnodes) {
  __shared__ float WTl[64 * LDSW];
  __shared__ float WTr[64 * LDSW];
  int t = threadIdx.x;
  // Stage W_l, W_r transposed into LDS: WT[n][k] = W[k][n]
#pragma unroll
  for (int c = 0; c < 16; ++c) {
    int idx = t + 256 * c;  // 4096 elements each
    int i = idx >> 6, j = idx & 63;
    WTl[j * LDSW + i] = Wl[idx];
    WTr[j * LDSW + i] = Wr[idx];
  }
  __syncthreads();

  int wave = t >> 5;
  int lane = t & 31;
  int m  = lane & 15;   // row within tile (A) / col within n-tile (B, C/D)
  int kh = lane >> 4;   // K-pair half select
  long r0 = ((long)blockIdx.x * 8 + wave) * 16;
  long row = r0 + m;
  long rowc = row < nnodes ? row : (long)(nnodes - 1);  // clamp: EXEC must stay full for WMMA
  float invd = 1.0f / fmaxf(deg[rowc], 1.0f);

  const float* sr = sum + rowc * 64;
  const float* xr = x + rowc * 64;

  v8f acc[4] = {v8f{}, v8f{}, v8f{}, v8f{}};

#pragma unroll
  for (int kc = 0; kc < 16; ++kc) {
    int k = kc * 4 + kh * 2;  // even -> 8B aligned
    v2f am = *(const v2f*)(sr + k);
    am = am * invd;                       // fused mean
    v2f ax = *(const v2f*)(xr + k);
#pragma unroll
    for (int nt = 0; nt < 4; ++nt) {
      v2f bl = *(const v2f*)&WTl[(nt * 16 + m) * LDSW + k];  // ds_load_b64
      acc[nt] = __builtin_amdgcn_wmma_f32_16x16x4_f32(
          false, am, false, bl, (short)0, acc[nt], false, false);
      v2f br = *(const v2f*)&WTr[(nt * 16 + m) * LDSW + k];
      acc[nt] = __builtin_amdgcn_wmma_f32_16x16x4_f32(
          false, ax, false, br, (short)0, acc[nt], false, false);
    }
  }

  // C/D layout: VGPR j -> row r0 + j + 8*kh, col m (+16 per n-tile)
  long base = r0 + 8 * kh;
#pragma unroll
  for (int j = 0; j < 8; ++j) {
    long rs = base + j;
    if (rs < nnodes) {
      float* o = out + rs * 64 + m;
      o[0]  = acc[0][j];
      o[16] = acc[1][j];
      o[32] = acc[2][j];
      o[48] = acc[3][j];
    }
  }
}

extern "C" void kernel_launch(void* const* d_in, const int* in_sizes, int n_in,
                              void* d_out, int out_size, void* d_ws, size_t ws_size,
                              hipStream_t stream) {
  const float* x      = (const float*)d_in[0];
  const long long* ei = (const long long*)d_in[1];  // int64 edge_index [2, E]
  const float* Wl     = (const float*)d_in[2];
  const float* Wr     = (const float*)d_in[3];
  float* out          = (float*)d_out;

  long nnodes = (long)in_sizes[0] / 64;
  long E      = (long)in_sizes[1] / 2;

  float* sum = (float*)d_ws;          // [nnodes*64]
  float* deg = sum + nnodes * 64;     // [nnodes]

  long zn = nnodes * 64 + nnodes;
  zero_ws_kernel<<<(int)((zn + 255) / 256), 256, 0, stream>>>(sum, zn);

  long st = E * 16;
  sage_scatter_kernel<<<(int)((st + 255) / 256), 256, 0, stream>>>(x, ei, sum, deg, E);

  int blocks = (int)((nnodes + 127) / 128);
  sage_gemm_kernel<<<blocks, 256, 0, stream>>>(x, Wl, Wr, sum, deg, out, (int)nnodes);
}